// ChamferDistanceLoss_80418967650489
// MI455X (gfx1250) — compile-verified
//
#include <hip/hip_runtime.h>

typedef __attribute__((ext_vector_type(2))) float v2f;
typedef __attribute__((ext_vector_type(8))) float v8f;

#define NBATCH     4
#define HW         76800          // 240*320
#define NTILE_T    4800           // HW/16
#define NC_REAL    257            // 256 centers + pad column
#define NC_PAD     272            // padded to 17*16
#define CT_TILES   17
#define BLK_PER_B  75
#define WAVES_PB   (BLK_PER_B * 8)   // 600 waves per batch -> 8 tiles/wave
#define DUMMY_C    1e18f

__device__ __forceinline__ float finf() { return __builtin_inff(); }

// ---------------- init workspace ----------------
__global__ void chamfer_init(unsigned* maxbits, float* term2, unsigned* colmin) {
    int i = blockIdx.x * blockDim.x + threadIdx.x;
    int stride = gridDim.x * blockDim.x;
    if (i < 2) maxbits[i] = 0u;
    if (i < NBATCH) term2[i] = 0.0f;
    for (int j = i; j < NBATCH * NC_PAD; j += stride) colmin[j] = 0x7F800000u; // +inf
}

// ---------------- masked max reduction (for pad_value) ----------------
__global__ __launch_bounds__(256) void chamfer_max(const float* __restrict__ target,
                                                   const float* __restrict__ centers,
                                                   const int* __restrict__ mask,
                                                   unsigned* maxbits) {
    int gid = blockIdx.x * blockDim.x + threadIdx.x;
    int stride = gridDim.x * blockDim.x;
    float lm = 0.0f;   // all candidates are uniform[0,1) >= 0
    for (int i = gid; i < NBATCH * HW; i += stride)
        lm = fmaxf(lm, (mask[i] != 0) ? target[i] : 0.0f);
    float lc = 0.0f;
    for (int i = gid; i < NBATCH * 256; i += stride)
        lc = fmaxf(lc, centers[i]);
    #pragma unroll
    for (int m = 1; m <= 16; m <<= 1) {
        lm = fmaxf(lm, __shfl_xor(lm, m, 32));
        lc = fmaxf(lc, __shfl_xor(lc, m, 32));
    }
    if ((threadIdx.x & 31) == 0) {
        atomicMax(&maxbits[0], __float_as_uint(lm));
        atomicMax(&maxbits[1], __float_as_uint(lc));
    }
}

// ---------------- WMMA main: bidirectional mins ----------------
// nct is CT_TILES passed at runtime so the compiler cannot fully unroll the
// center-tile loop (which previously blew past 256 VGPRs).
__global__ __launch_bounds__(256) void chamfer_main(const float* __restrict__ target,
                                                    const float* __restrict__ centers,
                                                    const int* __restrict__ mask,
                                                    const unsigned* __restrict__ maxbits,
                                                    float* term2, unsigned* colmin,
                                                    int nct) {
    __shared__ float    ldsC[NC_PAD];     // mod_centers, padded
    __shared__ unsigned ldsMin[NC_PAD];   // per-block running per-center min (float bits)

    const int b     = blockIdx.y;
    const int tid   = threadIdx.x;
    const int lane  = tid & 31;
    const int wave  = tid >> 5;
    const int col   = lane & 15;
    const int half8 = (lane >> 4) << 3;   // 0 for lanes 0-15, 8 for 16-31

    const float mt = __uint_as_float(maxbits[0]);
    const float mc = __uint_as_float(maxbits[1]);
    const float mx = fmaxf(mt, mc), mn = fminf(mt, mc);
    const float pad = mx + (mx - mn) + 1.0f;         // EPS = 1.0

    for (int i = tid; i < NC_PAD; i += 256) {
        ldsC[i]   = (i < 256) ? centers[b * 256 + i] : ((i == 256) ? pad : DUMMY_C);
        ldsMin[i] = 0x7F800000u;                     // +inf
    }
    __syncthreads();

    const int gw = blockIdx.x * 8 + wave;            // 0..599 within batch
    float t2sum = 0.0f;

    for (int t = gw; t < NTILE_T; t += WAVES_PB) {   // wave-uniform, exactly 8 iters
        int g = b * HW + t * 16 + col;

        // run-ahead prefetch of next tile (stride 600*16 elems); speculative, OOB-safe
        __builtin_prefetch(&target[g + WAVES_PB * 16], 0, 1);
        __builtin_prefetch(&mask[g + WAVES_PB * 16], 0, 1);

        float tv   = target[g];
        float tmod = (mask[g] != 0) ? tv : pad;
        v2f a; a.x = tmod; a.y = tmod;               // replicate into all K slots
        float tt = tmod * tmod;

        float tRow[8];
        #pragma unroll
        for (int v = 0; v < 8; ++v)                  // t^2 of row (v + half8)
            tRow[v] = __shfl(tt, v + half8, 32);

        float accD[8];
        #pragma unroll
        for (int v = 0; v < 8; ++v) accD[v] = finf();

        #pragma clang loop unroll_count(2)
        for (int ct = 0; ct < nct; ++ct) {           // runtime bound: stays rolled
            float c = ldsC[ct * 16 + col];
            v2f bb; bb.x = c; bb.y = c;
            float cs = c * c;
            v8f cz = {};
            // D = 4 * t[m] * c[n]  (all 4 K-slots populated identically)
            v8f D = __builtin_amdgcn_wmma_f32_16x16x4_f32(
                false, a, false, bb, (short)0, cz, false, false);
            float cm = finf();
            #pragma unroll
            for (int v = 0; v < 8; ++v) {
                float d2 = fmaf(-0.5f, D[v], tRow[v] + cs); // t^2 - 2tc + c^2
                accD[v] = fminf(accD[v], d2);
                cm = fminf(cm, d2);
            }
            // per-center min over this tile's 8 half-rows -> block LDS running min
            atomicMin(&ldsMin[ct * 16 + col], __float_as_uint(cm));
        }

        // term2: per-target min over all 272 centers, then sum over 16 targets
        float ssum = 0.0f;
        #pragma unroll
        for (int v = 0; v < 8; ++v) {
            float r = accD[v];
            r = fminf(r, __shfl_xor(r, 1, 32));
            r = fminf(r, __shfl_xor(r, 2, 32));
            r = fminf(r, __shfl_xor(r, 4, 32));
            r = fminf(r, __shfl_xor(r, 8, 32));      // min over N within each 16-half
            ssum += r;                               // row (v + half8) min
        }
        ssum += __shfl_xor(ssum, 16, 32);            // merge halves -> sum of 16 rows
        t2sum += ssum;
    }

    if (lane == 0) atomicAdd(&term2[b], t2sum);

    __syncthreads();
    for (int i = tid; i < NC_PAD; i += 256)          // one flush per block
        atomicMin(&colmin[b * NC_PAD + i], ldsMin[i]);
}

// ---------------- final: sum term1 + term2, mean over batch ----------------
__global__ __launch_bounds__(256) void chamfer_final(const unsigned* __restrict__ colmin,
                                                     const float* __restrict__ term2,
                                                     float* __restrict__ out) {
    __shared__ float red[256];
    float s = 0.0f;
    for (int i = threadIdx.x; i < NBATCH * NC_REAL; i += 256) {
        int b = i / NC_REAL, c = i % NC_REAL;
        s += __uint_as_float(colmin[b * NC_PAD + c]);
    }
    if (threadIdx.x < NBATCH) s += term2[threadIdx.x];
    red[threadIdx.x] = s;
    __syncthreads();
    for (int st = 128; st > 0; st >>= 1) {
        if (threadIdx.x < st) red[threadIdx.x] += red[threadIdx.x + st];
        __syncthreads();
    }
    if (threadIdx.x == 0) out[0] = red[0] * (1.0f / NBATCH);
}

extern "C" void kernel_launch(void* const* d_in, const int* in_sizes, int n_in,
                              void* d_out, int out_size, void* d_ws, size_t ws_size,
                              hipStream_t stream) {
    const float* target  = (const float*)d_in[0];  // (4,240,320) f32
    const float* centers = (const float*)d_in[1];  // (4,256) f32
    const int*   mask    = (const int*)d_in[2];    // (4,240,320) int (bool)
    float* out = (float*)d_out;

    unsigned char* ws = (unsigned char*)d_ws;
    unsigned* maxbits = (unsigned*)(ws + 0);       // 2 x u32
    float*    term2   = (float*)(ws + 8);          // 4 x f32
    unsigned* colmin  = (unsigned*)(ws + 32);      // 4*272 x u32

    chamfer_init <<<5, 256, 0, stream>>>(maxbits, term2, colmin);
    chamfer_max  <<<120, 256, 0, stream>>>(target, centers, mask, maxbits);
    chamfer_main <<<dim3(BLK_PER_B, NBATCH), 256, 0, stream>>>(target, centers, mask,
                                                               maxbits, term2, colmin,
                                                               CT_TILES);
    chamfer_final<<<1, 256, 0, stream>>>(colmin, term2, out);
}